// LocalDiskBlur_25262997635689
// MI455X (gfx1250) — compile-verified
//
#include <hip/hip_runtime.h>
#include <stdint.h>

#define KS    11
#define PADR  5
#define IW    512
#define IH    512
#define IHW   (IW * IH)
#define TX    32
#define TY    8
#define T     2                    /* pixels per thread in y */
#define NTHR  (TX * TY)
#define TILEY (TY * T)             /* 16 pixel rows per block */
#define LW    (TX + KS - 1)        /* 42 */
#define LH    (TILEY + KS - 1)     /* 26 */
#define CELLS (LW * LH)            /* 1092 */
#define NROWS (KS + T - 1)         /* 12 LDS rows touched per thread */
#define HALF  (KS / 2)             /* 5 symmetric pairs per row */

// ---- compile-time tap distances (fold to inline literals in the unrolled loop) ----
constexpr float csqrtf_ct(float x) {
  if (x <= 0.0f) return 0.0f;
  float g = x > 1.0f ? x : 1.0f;
  for (int i = 0; i < 48; ++i) g = 0.5f * (g + x / g);
  return g;
}
struct DistTab { float d[KS][KS]; };
constexpr DistTab makeDist() {
  DistTab t{};
  for (int i = 0; i < KS; ++i)
    for (int j = 0; j < KS; ++j) {
      float dy = (float)i - 5.0f, dx = (float)j - 5.0f;
      t.d[i][j] = csqrtf_ct(dy * dy + dx * dx);
    }
  return t;
}
constexpr DistTab DT = makeDist();

// ---- CDNA5 async global -> LDS copy (ASYNCcnt path, gfx1250) ----
// the LDS address comes from the VDST operand; generic shared pointers carry the LDS byte
// offset in their low 32 bits, so truncation yields the VDST operand.
__device__ __forceinline__ void async_copy_b32(const float* gsrc, float* ldst) {
  const unsigned lds_off = (unsigned)(unsigned long long)(uintptr_t)ldst;
  const unsigned long long gaddr = (unsigned long long)(uintptr_t)gsrc;
  asm volatile("global_load_async_to_lds_b32 %0, %1, off"
               :: "v"(lds_off), "v"(gaddr)
               : "memory");
}

__device__ __forceinline__ void wait_async_zero() {
#if defined(__has_builtin) && __has_builtin(__builtin_amdgcn_s_wait_asynccnt)
  __builtin_amdgcn_s_wait_asynccnt(0);
#else
  asm volatile("s_wait_asynccnt 0" ::: "memory");
#endif
}

__device__ __forceinline__ float clamp01(float x) {
  return __builtin_fminf(__builtin_fmaxf(x, 0.0f), 1.0f);
}

__global__ __launch_bounds__(NTHR)
void LocalDiskBlur_kernel(const float* __restrict__ img,
                          const float* __restrict__ mod,
                          float* __restrict__ out) {
  // Channel-interleaved tile: one ds_load_b128 per tap serves all 3 channels.
  __shared__ float4 sImg[CELLS];

  const int tx  = threadIdx.x;            // 0..31
  const int ty  = threadIdx.y;            // 0..7
  const int tid = ty * TX + tx;
  const int bx0 = blockIdx.x * TX;
  const int by0 = blockIdx.y * TILEY;

  // ---- stage 42x26 halo tile (replicate padding) via async copies ----
  for (int idx = tid; idx < CELLS; idx += NTHR) {
    const int cy = idx / LW;
    const int cx = idx - cy * LW;
    int gx = bx0 + cx - PADR;
    int gy = by0 + cy - PADR;
    gx = gx < 0 ? 0 : (gx > IW - 1 ? IW - 1 : gx);
    gy = gy < 0 ? 0 : (gy > IH - 1 ? IH - 1 : gy);
    const int gp = gy * IW + gx;
    float* cell = (float*)&sImg[idx];
    async_copy_b32(img + gp,           cell + 0);
    async_copy_b32(img + gp + IHW,     cell + 1);
    async_copy_b32(img + gp + 2 * IHW, cell + 2);
  }
  wait_async_zero();
  __syncthreads();

  // ---- per-pixel disk weights; each thread owns T consecutive pixel rows ----
  const int px  = bx0 + tx;
  const int py0 = by0 + ty * T;

  float coc5[T];
  float accR[T], accG[T], accB[T], ws[T];
#pragma unroll
  for (int t = 0; t < T; ++t) {
    const float m = mod[(py0 + t) * IW + px];
    coc5[t] = __builtin_fmaf(m, 0.5f, 0.5f);   // coc + 0.5
    accR[t] = accG[t] = accB[t] = ws[t] = 0.0f;
  }

  // Each LDS row is loaded once (11 independent b128 loads -> one DS wait),
  // mirror taps j and KS-1-j are pair-summed ONCE per row (they share the
  // same distance, hence the same weight for every pixel), and the pair sums
  // are reused by every owned pixel whose 11x11 window covers this row.
#pragma unroll
  for (int r = 0; r < NROWS; ++r) {
    float4 row[KS];
    const int rbase = (ty * T + r) * LW + tx;
#pragma unroll
    for (int j = 0; j < KS; ++j) row[j] = sImg[rbase + j];

    float psR[HALF], psG[HALF], psB[HALF];
#pragma unroll
    for (int j = 0; j < HALF; ++j) {
      psR[j] = row[j].x + row[KS - 1 - j].x;
      psG[j] = row[j].y + row[KS - 1 - j].y;
      psB[j] = row[j].z + row[KS - 1 - j].z;
    }
    const float cR = row[HALF].x, cG = row[HALF].y, cB = row[HALF].z;

#pragma unroll
    for (int t = 0; t < T; ++t) {
      const int ki = r - t;                    // compile-time after unroll
      if (ki >= 0 && ki < KS) {
#pragma unroll
        for (int j = 0; j < HALF; ++j) {
          const float w = clamp01(coc5[t] - DT.d[ki][j]);
          accR[t] = __builtin_fmaf(w, psR[j], accR[t]);
          accG[t] = __builtin_fmaf(w, psG[j], accG[t]);
          accB[t] = __builtin_fmaf(w, psB[j], accB[t]);
          ws[t]   = __builtin_fmaf(w, 2.0f,   ws[t]);   // pair contributes 2w
        }
        const float w = clamp01(coc5[t] - DT.d[ki][HALF]);
        accR[t] = __builtin_fmaf(w, cR, accR[t]);
        accG[t] = __builtin_fmaf(w, cG, accG[t]);
        accB[t] = __builtin_fmaf(w, cB, accB[t]);
        ws[t]  += w;
      }
    }
  }

#pragma unroll
  for (int t = 0; t < T; ++t) {
    const float inv = 1.0f / (ws[t] + 1e-8f);
    const int   op  = (py0 + t) * IW + px;
    out[op]           = accR[t] * inv;
    out[op + IHW]     = accG[t] * inv;
    out[op + 2 * IHW] = accB[t] * inv;
  }
}

extern "C" void kernel_launch(void* const* d_in, const int* in_sizes, int n_in,
                              void* d_out, int out_size, void* d_ws, size_t ws_size,
                              hipStream_t stream) {
  (void)in_sizes; (void)n_in; (void)out_size; (void)d_ws; (void)ws_size;
  const float* img = (const float*)d_in[0];   // (1,3,512,512) f32
  const float* mod = (const float*)d_in[1];   // (512,512)     f32
  float*       out = (float*)d_out;           // (1,3,512,512) f32

  dim3 block(TX, TY, 1);
  dim3 grid(IW / TX, IH / TILEY, 1);          // 16 x 32 = 512 blocks
  hipLaunchKernelGGL(LocalDiskBlur_kernel, grid, block, 0, stream, img, mod, out);
}